// fast_BEV_10488310137173
// MI455X (gfx1250) — compile-verified
//
#include <hip/hip_runtime.h>
#include <hip/hip_bf16.h>
#include <stddef.h>

// ---------------- problem constants (from reference) ----------------
#define NXv    200
#define NYv    200
#define NZv    4
#define Bc     2
#define NCAMc  6
#define Cc     256
#define HFc    64
#define WFc    176
#define OUT_Cc 80
#define PIXc   (HFc * WFc)        // 11264
#define Vv     (NXv * NYv * NZv)  // 160000
#define HWc    (NXv * NYv)        // 40000
#define BN_EPSf 1e-5f

typedef __attribute__((ext_vector_type(2))) float v2f;
typedef __attribute__((ext_vector_type(8))) float v8f;

// ---------------- kernel 1: fold projection matrices ----------------
// For each (b, cam): p2i = M3 * g + t3 ;  u = p0/p2 + tu ; v = p1/p2 + tv
// M3 = proj[:, :3] @ lr^T ; t3 = proj[:,3] - M3 @ lt ; proj = (img_r @ l2i)[:3]
__global__ void proj_mats_kernel(const float* __restrict__ l2i,
                                 const float* __restrict__ iaug,
                                 const float* __restrict__ laug,
                                 float* __restrict__ pm) {
    int t = threadIdx.x;
    if (t >= Bc * NCAMc) return;
    int b = t / NCAMc;
    const float* L = l2i  + t * 16;
    const float* A = iaug + t * 16;
    const float* G = laug + b * 16;
    float proj[3][4];
    for (int i = 0; i < 3; ++i)
        for (int j = 0; j < 4; ++j) {
            float s = 0.f;                     // img_r has column 3 zeroed
            for (int k = 0; k < 3; ++k) s += A[i * 4 + k] * L[k * 4 + j];
            proj[i][j] = s;
        }
    float M[3][3];
    for (int i = 0; i < 3; ++i)
        for (int j = 0; j < 3; ++j)            // lr^T[k][j] = laug[j][k]
            M[i][j] = proj[i][0] * G[j * 4 + 0] + proj[i][1] * G[j * 4 + 1] +
                      proj[i][2] * G[j * 4 + 2];
    float lt0 = G[3], lt1 = G[7], lt2 = G[11];
    float* P = pm + t * 16;
    for (int i = 0; i < 3; ++i) {
        P[i * 3 + 0] = M[i][0]; P[i * 3 + 1] = M[i][1]; P[i * 3 + 2] = M[i][2];
        P[9 + i] = proj[i][3] - (M[i][0] * lt0 + M[i][1] * lt1 + M[i][2] * lt2);
    }
    P[12] = A[3];   // img_t.u
    P[13] = A[7];   // img_t.v
}

// ---------------- kernel 2: per-voxel winning camera index ----------------
__global__ __launch_bounds__(256) void voxel_proj_kernel(
    const float* __restrict__ pm, int* __restrict__ idx) {
    int t = blockIdx.x * 256 + threadIdx.x;
    if (t >= Bc * Vv) return;
    int b = t / Vv, v = t - b * Vv;
    int x = v / (NYv * NZv);
    int r = v - x * (NYv * NZv);
    int y = r / NZv, z = r - y * NZv;
    float gx = x * 0.5f - 50.0f;
    float gy = y * 0.5f - 50.0f;
    float gz = z * 1.5f - 4.0f;
    int win = -1;
    for (int n = 0; n < NCAMc; ++n) {          // last valid camera wins
        const float* P = pm + (b * NCAMc + n) * 16;
        float pz = P[6] * gx + P[7] * gy + P[8] * gz + P[11];
        if (pz > 0.f) {
            float u  = (P[0] * gx + P[1] * gy + P[2] * gz + P[9])  / pz + P[12];
            float vv = (P[3] * gx + P[4] * gy + P[5] * gz + P[10]) / pz + P[13];
            int uf = (int)rintf(u  * 0.25f);   // STRIDE=4, round half-to-even
            int vf = (int)rintf(vv * 0.25f);
            if (uf >= 0 && uf < WFc && vf >= 0 && vf < HFc)
                win = n * PIXc + vf * WFc + uf;
        }
    }
    idx[t] = win;
}

// ---------------- kernel 3: conv_w (O,C) -> K-major (C,O) ----------------
__global__ void convw_transpose_kernel(const float* __restrict__ w,
                                       float* __restrict__ wt) {
    int t = blockIdx.x * 256 + threadIdx.x;
    if (t < Cc * OUT_Cc) {
        int c = t / OUT_Cc, o = t - c * OUT_Cc;
        wt[t] = w[o * Cc + c];                 // wt[c*80+o], coalesced write
    }
}

// ---------------- kernel 4: feature transpose (B,N,C,H,W)->(B,N,H,W,C) ----
__global__ __launch_bounds__(256) void feat_transpose_kernel(
    const float* __restrict__ src, float* __restrict__ dst) {
    __shared__ float tile[32][33];
    int img = blockIdx.z;
    size_t sb = (size_t)img * Cc * PIXc;
    size_t db = (size_t)img * PIXc * Cc;
    int p0 = blockIdx.x * 32, c0 = blockIdx.y * 32;
#pragma unroll
    for (int j = 0; j < 4; ++j) {
        int c = c0 + threadIdx.y + j * 8;
        tile[threadIdx.y + j * 8][threadIdx.x] =
            src[sb + (size_t)c * PIXc + p0 + threadIdx.x];
    }
    __syncthreads();
#pragma unroll
    for (int j = 0; j < 4; ++j) {
        int p = p0 + threadIdx.y + j * 8;
        dst[db + (size_t)p * Cc + c0 + threadIdx.x] =
            tile[threadIdx.x][threadIdx.y + j * 8];
    }
}

// ---------------- kernel 5: gather + weighted-z + WMMA 1x1 conv ----------
// Block: 256 threads (8 waves), 32 BEV cells. LDS bev tile padded to 260
// floats/row so the 16x4 f32 A-fragment b64 loads hit 64 distinct LDS banks.
// GEMM: 5 waves, one per 16-wide N-tile; each wave carries TWO accumulators
// (M-tiles 0 and 1) and shares the B fragment between the two WMMAs.
__global__ __launch_bounds__(256) void bev_gemm_kernel(
    const float* __restrict__ featT, const float* __restrict__ feat,
    const int* __restrict__ idx, const float* __restrict__ points,
    const float* __restrict__ cwT, const float* __restrict__ conv_b,
    float* __restrict__ out, int transposed) {
    __shared__ float bev[32][Cc + 4];          // 33,280 B
    __shared__ float ytile[OUT_Cc][33];        // 10,560 B
    int t = threadIdx.x;
    int b = blockIdx.y;
    int cell0 = blockIdx.x * 32;

    // ---- gather phase: thread t owns channel c = t ----
    int c = t;
    for (int i = 0; i < 32; ++i) {
        int cell = cell0 + i;
        int x = cell / NYv, yy = cell - x * NYv;
        float acc = 0.f;
#pragma unroll
        for (int z = 0; z < NZv; ++z) {
            int v  = (x * NYv + yy) * NZv + z;
            int id = idx[b * Vv + v];          // uniform across block
            if (id >= 0) {
                float wgt = points[((b * NZv + z) * NXv + x) * NYv + yy];
                float fv;
                if (transposed) {              // contiguous 1KB burst over c
                    fv = featT[((size_t)(b * NCAMc * PIXc) + id) * Cc + c];
                } else {                       // fallback: strided gather
                    int n = id / PIXc;
                    int p = id - n * PIXc;
                    fv = feat[((size_t)((b * NCAMc + n) * Cc + c)) * PIXc + p];
                }
                acc += wgt * fv;
            }
        }
        bev[i][c] = acc;
    }
    __syncthreads();

    // ---- GEMM phase: y[32 x 80] = bev[32 x 256] * cwT[256 x 80] + bias ----
    int wave = t >> 5, lane = t & 31;
    int lo = lane & 15, hi = lane >> 4;
    if (wave < 5) {                            // wave = N-tile index
        int n0 = wave * 16;
        v8f acc0, acc1;
        float bias = conv_b[n0 + lo];
#pragma unroll
        for (int r = 0; r < 8; ++r) { acc0[r] = bias; acc1[r] = bias; }
#pragma unroll 4
        for (int k = 0; k < Cc; k += 4) {
            // B fragment (4x16 f32): lane=N, VGPR0/1 = K pair — shared by
            // both M-tiles (register reuse, no second load).
            v2f bb;
            bb.x = cwT[(k + 2 * hi)     * OUT_Cc + n0 + lo];
            bb.y = cwT[(k + 2 * hi + 1) * OUT_Cc + n0 + lo];
            // A fragments (16x4 f32): lane=M, VGPR0/1 = K pair, hi lanes K+2
            v2f a0, a1;
            a0.x = bev[lo][k + 2 * hi];
            a0.y = bev[lo][k + 2 * hi + 1];
            a1.x = bev[16 + lo][k + 2 * hi];
            a1.y = bev[16 + lo][k + 2 * hi + 1];
            acc0 = __builtin_amdgcn_wmma_f32_16x16x4_f32(
                false, a0, false, bb, (short)0, acc0, false, false);
            acc1 = __builtin_amdgcn_wmma_f32_16x16x4_f32(
                false, a1, false, bb, (short)0, acc1, false, false);
        }
        // D layout: VGPR r -> M = r + 8*hi, N = lo
#pragma unroll
        for (int r = 0; r < 8; ++r) {
            ytile[n0 + lo][r + 8 * hi]      = acc0[r];
            ytile[n0 + lo][16 + r + 8 * hi] = acc1[r];
        }
    }
    __syncthreads();

    // ---- coalesced store of pre-BN y ----
    for (int e = t; e < 32 * OUT_Cc; e += 256) {
        int o = e >> 5, i = e & 31;
        out[((size_t)(b * OUT_Cc + o)) * HWc + cell0 + i] = ytile[o][i];
    }
}

// ---------------- kernel 6: deterministic per-channel BN reduction --------
__global__ __launch_bounds__(256) void bn_reduce_kernel(
    const float* __restrict__ y, float* __restrict__ sums) {
    __shared__ float s1[256], s2[256];
    int o = blockIdx.x, t = threadIdx.x;
    float a = 0.f, b = 0.f;
    for (int e = t; e < Bc * HWc; e += 256) {
        int bb = e / HWc, p = e - bb * HWc;
        float v = y[((size_t)(bb * OUT_Cc + o)) * HWc + p];
        a += v; b += v * v;
    }
    s1[t] = a; s2[t] = b;
    __syncthreads();
    for (int s = 128; s > 0; s >>= 1) {
        if (t < s) { s1[t] += s1[t + s]; s2[t] += s2[t + s]; }
        __syncthreads();
    }
    if (t == 0) { sums[o] = s1[0]; sums[OUT_Cc + o] = s2[0]; }
}

// ---------------- kernel 7: BN scale/shift coefficients -------------------
__global__ void bn_finalize_kernel(const float* __restrict__ sums,
                                   const float* __restrict__ gamma,
                                   const float* __restrict__ beta,
                                   float* __restrict__ ab) {
    int o = threadIdx.x;
    if (o < OUT_Cc) {
        float n = (float)(Bc * HWc);
        float mean = sums[o] / n;
        float var  = sums[OUT_Cc + o] / n - mean * mean;
        float rstd = rsqrtf(var + BN_EPSf);
        float a = gamma[o] * rstd;
        ab[o] = a;
        ab[OUT_Cc + o] = beta[o] - mean * a;
    }
}

// ---------------- kernel 8: fused normalize + ReLU in place ---------------
__global__ __launch_bounds__(256) void bn_apply_kernel(
    float* __restrict__ y, const float* __restrict__ ab) {
    int t = blockIdx.x * 256 + threadIdx.x;
    if (t < Bc * OUT_Cc * HWc) {
        int o = (t / HWc) % OUT_Cc;
        float v = y[t] * ab[o] + ab[OUT_Cc + o];
        y[t] = fmaxf(v, 0.f);
    }
}

// ---------------- host launcher ----------------
extern "C" void kernel_launch(void* const* d_in, const int* in_sizes, int n_in,
                              void* d_out, int out_size, void* d_ws,
                              size_t ws_size, hipStream_t stream) {
    const float* feat   = (const float*)d_in[0];
    const float* points = (const float*)d_in[1];
    const float* l2i    = (const float*)d_in[2];
    const float* iaug   = (const float*)d_in[3];
    const float* laug   = (const float*)d_in[4];
    const float* convw  = (const float*)d_in[5];
    const float* convb  = (const float*)d_in[6];
    const float* gamma  = (const float*)d_in[7];
    const float* beta   = (const float*)d_in[8];
    float* out = (float*)d_out;
    char* ws = (char*)d_ws;

    const size_t off_proj = 0;                                   // 768 B
    const size_t off_idx  = 1024;                                // 1.28 MB
    const size_t off_cwT  = off_idx + (size_t)Bc * Vv * 4;       // 80 KB
    const size_t off_sums = off_cwT + (size_t)Cc * OUT_Cc * 4;   // 640 B
    const size_t off_bn   = off_sums + 2 * OUT_Cc * 4;           // 640 B
    size_t off_featT = (off_bn + 2 * OUT_Cc * 4 + 255) & ~(size_t)255;
    const size_t featT_bytes = (size_t)Bc * NCAMc * PIXc * Cc * 4; // 138 MB
    int transposed = (ws_size >= off_featT + featT_bytes) ? 1 : 0;

    float* pm    = (float*)(ws + off_proj);
    int*   idx   = (int*)(ws + off_idx);
    float* cwT   = (float*)(ws + off_cwT);
    float* sums  = (float*)(ws + off_sums);
    float* ab    = (float*)(ws + off_bn);
    float* featT = (float*)(ws + off_featT);

    proj_mats_kernel<<<1, 16, 0, stream>>>(l2i, iaug, laug, pm);
    voxel_proj_kernel<<<(Bc * Vv + 255) / 256, 256, 0, stream>>>(pm, idx);
    convw_transpose_kernel<<<(Cc * OUT_Cc + 255) / 256, 256, 0, stream>>>(convw, cwT);
    if (transposed)
        feat_transpose_kernel<<<dim3(PIXc / 32, Cc / 32, Bc * NCAMc),
                                dim3(32, 8), 0, stream>>>(feat, featT);
    bev_gemm_kernel<<<dim3(HWc / 32, Bc), 256, 0, stream>>>(
        featT, feat, idx, points, cwT, convb, out, transposed);
    bn_reduce_kernel<<<OUT_Cc, 256, 0, stream>>>(out, sums);
    bn_finalize_kernel<<<1, 128, 0, stream>>>(sums, gamma, beta, ab);
    bn_apply_kernel<<<(Bc * OUT_Cc * HWc + 255) / 256, 256, 0, stream>>>(out, ab);
}